// DRMLLayer_1932735283346
// MI455X (gfx1250) — compile-verified
//
#include <hip/hip_runtime.h>
#include <hip/hip_bf16.h>

// ---------------------------------------------------------------------------
// CDNA5 (gfx1250) implementation of the DRML reference pipeline.
// All conv / FC stages run as implicit-GEMM on V_WMMA_F32_16X16X4_F32
// (full fp32 precision, matching the fp32 reference). Weights are staged
// through LDS in chunks and read back as conflict-free ds_load_b64; the
// K-step loop is unrolled 4x so loads for 4 WMMA steps are clustered ahead
// of the serialized accumulator chain.
// ---------------------------------------------------------------------------

typedef __attribute__((ext_vector_type(2))) float v2f;
typedef __attribute__((ext_vector_type(8))) float v8f;

// ---------------------------------------------------------------------------
// Generic implicit-GEMM convolution using fp32 WMMA.
//   in  : (Bc, CIN, Hin, Win)  fp32
//   w   : (Cout, CIN, KH, KW)  fp32 (row-major == M x K with K = CIN*KH*KW)
//   out : (Bc, Cout, Hout, Wout)
// One wave computes a 16(M) x 16(N) output tile; K consumed 4/step via WMMA.
// Block = 128 threads = 4 wave32s sharing one M-tile -> LDS weight staging
// amortizes across the 4 waves.
// ---------------------------------------------------------------------------
template <int CIN, int KHH, int KWW>
__device__ __forceinline__ float conv_load_patch(const float* __restrict__ in,
                                                 int bb, int iy0, int ix0,
                                                 int Hin, int Win, int k) {
    constexpr int KHW = KHH * KWW;
    int c  = k / KHW;          // constant divisors -> mul/shift
    int r  = k % KHW;
    int ky = r / KWW;
    int kx = r % KWW;
    int iy = iy0 + ky;
    int ix = ix0 + kx;
    if (iy < 0 || iy >= Hin || ix < 0 || ix >= Win) return 0.0f;
    return in[((long)(bb * CIN + c) * Hin + iy) * Win + ix];
}

template <int CIN, int KHH, int KWW, int STRIDE, int PAD>
__global__ void __launch_bounds__(128)
conv_wmma_kernel(const float* __restrict__ in,
                 const float* __restrict__ w,
                 const float* __restrict__ bias,
                 float* __restrict__ out,
                 int Bc, int Hin, int Win,
                 int Cout, int Hout, int Wout) {
    constexpr int K   = CIN * KHH * KWW;
    constexpr int K4  = ((K + 3) / 4) * 4;      // K padded to WMMA step
    constexpr int KC  = 512;                    // k-chunk staged in LDS
    constexpr int KCS = KC + 4;                 // padded stride: 16 lanes ->
                                                // 16 distinct bank pairs
    __shared__ float sw[16 * KCS];              // 33,024 B

    const int lane   = threadIdx.x & 31;
    const int waveid = threadIdx.x >> 5;
    const int col    = lane & 15;               // N column inside tile
    const int krow   = (lane < 16) ? 0 : 2;     // K pair this lane supplies
    const int cout0  = blockIdx.y * 16;         // M tile origin

    const long Ntot = (long)Bc * Hout * Wout;
    const long n0   = ((long)blockIdx.x * 4 + waveid) * 16;

    long n = n0 + col;
    const bool valid = (n < Ntot);
    if (!valid) n = Ntot - 1;                   // clamp; store is masked

    const int hw  = Hout * Wout;
    const int bb  = (int)(n / hw);
    const int rem = (int)(n % hw);
    const int oy  = rem / Wout;
    const int ox  = rem % Wout;
    const int iy0 = oy * STRIDE - PAD;
    const int ix0 = ox * STRIDE - PAD;

    v8f acc = {};
    for (int chunk0 = 0; chunk0 < K4; chunk0 += KC) {
        const int kc_len = (K4 - chunk0 < KC) ? (K4 - chunk0) : KC; // mult of 4

        __syncthreads();  // previous chunk fully consumed before overwrite
        // Cooperative stage: 16 rows of the M-tile's weights, zero-padded tail.
        for (int m = 0; m < 16; ++m) {
            const float* __restrict__ wrow = w + (long)(cout0 + m) * K + chunk0;
            for (int kk = threadIdx.x; kk < kc_len; kk += 128) {
                const int k = chunk0 + kk;
                sw[m * KCS + kk] = (k < K) ? wrow[kk] : 0.0f;
            }
        }
        __syncthreads();

        const int steps = kc_len >> 2;
        #pragma unroll 4
        for (int s = 0; s < steps; ++s) {
            const int kk0 = s * 4 + krow;
            // A 16x4: lane m=col supplies K=kk0, kk0+1  (ds_load_b64)
            const v2f Av = *(const v2f*)(sw + col * KCS + kk0);
            // B 4x16: lane n=col supplies rows kk0, kk0+1 (im2col gather)
            const int k0 = chunk0 + kk0;
            float b0 = 0.0f, b1 = 0.0f;
            if (k0 < K)
                b0 = conv_load_patch<CIN, KHH, KWW>(in, bb, iy0, ix0, Hin, Win, k0);
            if (k0 + 1 < K)
                b1 = conv_load_patch<CIN, KHH, KWW>(in, bb, iy0, ix0, Hin, Win, k0 + 1);
            const v2f Bv = {b0, b1};
            acc = __builtin_amdgcn_wmma_f32_16x16x4_f32(
                /*neg_a=*/false, Av, /*neg_b=*/false, Bv,
                /*c_mod=*/(short)0, acc, /*reuse_a=*/false, /*reuse_b=*/false);
        }
    }

    // C/D layout: VGPR j -> row j (lanes 0-15) or row j+8 (lanes 16-31)
    const int mbase = (lane < 16) ? 0 : 8;
    if (valid) {
        #pragma unroll
        for (int j = 0; j < 8; ++j) {
            const int m = cout0 + mbase + j;
            out[((long)(bb * Cout + m) * Hout + oy) * Wout + ox] = acc[j] + bias[m];
        }
    }
}

// ---------------------------------------------------------------------------
// Region statistics: per (region rr=ri*4+ci, channel c) mean & 1/sqrt(var+eps)
// over B(64) x 40 x 40 = 102400 elements of h1 (64,32,160,160).
// ---------------------------------------------------------------------------
__global__ void region_stats_kernel(const float* __restrict__ h,
                                    float* __restrict__ stats) {
    const int id = blockIdx.x;          // 0..511 = rr*32 + c
    const int rr = id >> 5;
    const int c  = id & 31;
    const int ri = rr >> 2;
    const int ci = rr & 3;
    const int tid = threadIdx.x;

    float s = 0.0f, s2 = 0.0f;
    for (int i = tid; i < 64 * 1600; i += blockDim.x) {
        const int bb = i / 1600;
        const int p  = i % 1600;
        const int y  = p / 40;
        const int x  = p % 40;
        const float v =
            h[((long)(bb * 32 + c) * 160 + (ri * 40 + y)) * 160 + (ci * 40 + x)];
        s += v;
        s2 += v * v;
    }
    __shared__ float sh[256];
    __shared__ float sh2[256];
    sh[tid] = s;
    sh2[tid] = s2;
    __syncthreads();
    for (int off = 128; off > 0; off >>= 1) {
        if (tid < off) {
            sh[tid] += sh[tid + off];
            sh2[tid] += sh2[tid + off];
        }
        __syncthreads();
    }
    if (tid == 0) {
        const float invN = 1.0f / 102400.0f;
        const float mean = sh[0] * invN;
        const float var  = sh2[0] * invN - mean * mean;
        stats[id]       = mean;
        stats[512 + id] = rsqrtf(var + 1e-5f);
    }
}

// ---------------------------------------------------------------------------
// Normalize + PReLU, scattering h1 (B,C,160,160) into region-major layout
// r (16, 64, 32, 40, 40) so rconv becomes a plain batch-1024 padded conv.
// ---------------------------------------------------------------------------
__global__ void region_norm_kernel(const float* __restrict__ h,
                                   const float* __restrict__ stats,
                                   const float* __restrict__ gamma,
                                   const float* __restrict__ beta,
                                   const float* __restrict__ pa,
                                   float* __restrict__ r) {
    const float a = pa[0];
    const long total = 16L * 64 * 32 * 1600;   // 52,428,800
    for (long o = (long)blockIdx.x * blockDim.x + threadIdx.x; o < total;
         o += (long)gridDim.x * blockDim.x) {
        int t = (int)o;
        const int x = t % 40; t /= 40;
        const int y = t % 40; t /= 40;
        const int c = t % 32; t /= 32;
        const int bb = t % 64;
        const int rr = t / 64;
        const int ri = rr >> 2;
        const int ci = rr & 3;
        float v =
            h[((long)(bb * 32 + c) * 160 + (ri * 40 + y)) * 160 + (ci * 40 + x)];
        const int sid = rr * 32 + c;
        v = (v - stats[sid]) * stats[512 + sid] * gamma[c] + beta[c];
        r[o] = (v > 0.0f) ? v : a * v;
    }
}

// ---------------------------------------------------------------------------
// Reassemble region-major rconv output (16,64,32,40,40) + 2x2 maxpool
// -> (64,32,80,80). Region boundaries align with the pool grid (40 is even).
// ---------------------------------------------------------------------------
__global__ void pool_kernel(const float* __restrict__ rc,
                            float* __restrict__ out) {
    const long total = 64L * 32 * 80 * 80;     // 13,107,200
    for (long o = (long)blockIdx.x * blockDim.x + threadIdx.x; o < total;
         o += (long)gridDim.x * blockDim.x) {
        int t = (int)o;
        const int px = t % 80; t /= 80;
        const int py = t % 80; t /= 80;
        const int c  = t % 32;
        const int bb = t / 32;
        const int oy = py * 2, ox = px * 2;
        const int ri = oy / 40, ly = oy % 40;
        const int ci = ox / 40, lx = ox % 40;
        const int rr = ri * 4 + ci;
        const float* base =
            rc + ((long)((rr * 64 + bb) * 32 + c)) * 1600 + ly * 40 + lx;
        const float m = fmaxf(fmaxf(base[0], base[1]), fmaxf(base[40], base[41]));
        out[o] = m;
    }
}

// ---------------------------------------------------------------------------
// Host-side launch sequence.
// Workspace layout (floats): buf0[52,428,800] | buf1[52,428,800] | stats[1024]
//   total ~419.4 MB + 4 KB.
// ---------------------------------------------------------------------------
static inline dim3 conv_grid(long Ntot, int Cout) {
    const long ntiles = (Ntot + 15) / 16;
    return dim3((unsigned)((ntiles + 3) / 4), (unsigned)(Cout / 16), 1);
}

extern "C" void kernel_launch(void* const* d_in, const int* in_sizes, int n_in,
                              void* d_out, int out_size, void* d_ws, size_t ws_size,
                              hipStream_t stream) {
    const float* x     = (const float*)d_in[0];
    const float* c1w   = (const float*)d_in[1];
    const float* c1b   = (const float*)d_in[2];
    const float* gamma = (const float*)d_in[3];
    const float* beta  = (const float*)d_in[4];
    const float* pa    = (const float*)d_in[5];
    const float* rw    = (const float*)d_in[6];
    const float* rb    = (const float*)d_in[7];
    const float* c4w   = (const float*)d_in[8];
    const float* c4b   = (const float*)d_in[9];
    const float* c5w   = (const float*)d_in[10];
    const float* c5b   = (const float*)d_in[11];
    const float* c6w   = (const float*)d_in[12];
    const float* c6b   = (const float*)d_in[13];
    const float* c7w   = (const float*)d_in[14];
    const float* c7b   = (const float*)d_in[15];
    const float* c8w   = (const float*)d_in[16];
    const float* c8b   = (const float*)d_in[17];
    const float* fcw   = (const float*)d_in[18];
    const float* fcb   = (const float*)d_in[19];

    float* buf0  = (float*)d_ws;
    float* buf1  = buf0 + 52428800L;
    float* stats = buf1 + 52428800L;
    float* out   = (float*)d_out;

    // 1) conv1: (64,3,170,170) -> (64,32,160,160)   [buf0]
    conv_wmma_kernel<3, 11, 11, 1, 0>
        <<<conv_grid(64L * 160 * 160, 32), 128, 0, stream>>>(
            x, c1w, c1b, buf0, 64, 170, 170, 32, 160, 160);

    // 2) per-(region,channel) mean / invstd                [stats]
    region_stats_kernel<<<512, 256, 0, stream>>>(buf0, stats);

    // 3) normalize + PReLU -> region-major (16,64,32,40,40) [buf1]
    region_norm_kernel<<<4096, 256, 0, stream>>>(buf0, stats, gamma, beta, pa, buf1);

    // 4) region conv 3x3 pad1, batch=1024 images (32,40,40) [buf0]
    conv_wmma_kernel<32, 3, 3, 1, 1>
        <<<conv_grid(1024L * 40 * 40, 32), 128, 0, stream>>>(
            buf1, rw, rb, buf0, 1024, 40, 40, 32, 40, 40);

    // 5) reassemble + maxpool 2x2 -> (64,32,80,80)          [buf1]
    pool_kernel<<<4096, 256, 0, stream>>>(buf0, buf1);

    // 6) conv4 8x8: (64,32,80,80) -> (64,16,73,73)          [buf0]
    conv_wmma_kernel<32, 8, 8, 1, 0>
        <<<conv_grid(64L * 73 * 73, 16), 128, 0, stream>>>(
            buf1, c4w, c4b, buf0, 64, 80, 80, 16, 73, 73);

    // 7) conv5 8x8 s2: -> (64,16,33,33)                     [buf1]
    conv_wmma_kernel<16, 8, 8, 2, 0>
        <<<conv_grid(64L * 33 * 33, 16), 128, 0, stream>>>(
            buf0, c5w, c5b, buf1, 64, 73, 73, 16, 33, 33);

    // 8) conv6 6x6: -> (64,16,28,28)                        [buf0]
    conv_wmma_kernel<16, 6, 6, 1, 0>
        <<<conv_grid(64L * 28 * 28, 16), 128, 0, stream>>>(
            buf1, c6w, c6b, buf0, 64, 33, 33, 16, 28, 28);

    // 9) conv7 5x5: -> (64,16,24,24)                        [buf1]
    conv_wmma_kernel<16, 5, 5, 1, 0>
        <<<conv_grid(64L * 24 * 24, 16), 128, 0, stream>>>(
            buf0, c7w, c7b, buf1, 64, 28, 28, 16, 24, 24);

    // 10) conv8 3x3 s3: -> (64,16,8,8)                      [buf0]
    conv_wmma_kernel<16, 3, 3, 3, 0>
        <<<conv_grid(64L * 8 * 8, 16), 128, 0, stream>>>(
            buf1, c8w, c8b, buf0, 64, 24, 24, 16, 8, 8);

    // 11) FC 1024->512 as 1x1 conv over (64,1024,1,1) -> d_out (64,512)
    conv_wmma_kernel<1024, 1, 1, 1, 0>
        <<<conv_grid(64L, 512), 128, 0, stream>>>(
            buf0, fcw, fcb, out, 64, 1, 1, 512, 1, 1);
}